// Predictor_13082470383918
// MI455X (gfx1250) — compile-verified
//
#include <hip/hip_runtime.h>

#define N_ENT   50000
#define RANKS   3
#define NSTEP   3
#define NOP     12
#define NNZE    100000
#define BATCH   64
#define THRV    1e-20f
#define PER_RANK (N_ENT * BATCH)          /* 3,200,000 */
#define TOTAL    (RANKS * PER_RANK)       /* 9,600,000 */

typedef __attribute__((ext_vector_type(2))) float v2f;
typedef __attribute__((ext_vector_type(8))) float v8f;

__device__ __forceinline__ void atomAddF(float* p, float v) {
    unsafeAtomicAdd(p, v);   // lowers to global_atomic_add_f32 (L2-scope RMW)
}

__global__ void k_zero(float* __restrict__ p, int n) {
    int i = blockIdx.x * blockDim.x + threadIdx.x;
    if (i < n) p[i] = 0.0f;
}

// mem[r][tt[b]][b] = 1
__global__ void k_onehot(float* __restrict__ mem, const int* __restrict__ tt) {
    int t = threadIdx.x;                 // 0..191
    int r = t >> 6, b = t & 63;
    int n = tt[b];
    mem[((size_t)(r * N_ENT + n) << 6) + b] = 1.0f;
}

// s[r][n][b] = mem[r][n][b] * att[r][step][b][NOP]   (self-loop term)
__global__ void k_selfloop(float* __restrict__ s, const float* __restrict__ mem,
                           const float* __restrict__ att, int step) {
    int r = blockIdx.y;
    int i = blockIdx.x * blockDim.x + threadIdx.x;    // 0..PER_RANK-1 (exact)
    int b = i & 63;
    float w = att[((r * NSTEP + step) * BATCH + b) * (NOP + 1) + NOP];
    size_t idx = (size_t)r * PER_RANK + i;
    s[idx] = mem[idx] * w;
}

// s[r][row][b] += att[r][step][b][o] * val * mem[r][col][b]
__global__ void k_edges(float* __restrict__ s, const float* __restrict__ mem,
                        const float* __restrict__ att,
                        const int* __restrict__ rows, const int* __restrict__ cols,
                        const float* __restrict__ vals, int step) {
    __shared__ float watt[RANKS * BATCH];
    int o   = blockIdx.y;
    int tid = threadIdx.x;
    if (tid < RANKS * BATCH) {
        int rr = tid >> 6, bb = tid & 63;
        watt[tid] = att[((rr * NSTEP + step) * BATCH + bb) * (NOP + 1) + o];
    }
    __syncthreads();
    int i = blockIdx.x * blockDim.x + tid;            // covers NNZE*64 exactly
    int e = i >> 6, b = i & 63;
    int eo  = o * NNZE + e;
    int row = rows[eo], col = cols[eo];
    float val = vals[eo];
#pragma unroll
    for (int r = 0; r < RANKS; ++r) {
        float m = mem[((size_t)(r * N_ENT + col) << 6) + b];
        if (m != 0.0f) {
            atomAddF(&s[((size_t)(r * N_ENT + row) << 6) + b],
                     watt[(r << 6) + b] * val * m);
        }
    }
}

// norm[r][b] = sum_n |s[r][n][b]| via V_WMMA_F32_16X16X4_F32 (A = ones 16x4).
// D[i][j] = sum_k B[k][j]: exact f32 column sums. Wave w owns columns w*16..w*16+15.
// B layout (ISA 7.12.2, 32-bit): lanes 0-15 -> N=lane, K={0,1}; lanes 16-31 -> N=lane-16, K={2,3}.
__global__ __launch_bounds__(128) void k_norm_wmma(float* __restrict__ norm,
                                                   const float* __restrict__ s) {
    int r    = blockIdx.y;
    int lane = threadIdx.x & 31;
    int wave = threadIdx.x >> 5;                      // 0..3
    int col  = (wave << 4) + (lane & 15);             // batch column 0..63
    int kA   = (lane < 16) ? 0 : 2;
    int n0   = blockIdx.x * 400;                      // 125 blocks * 400 = 50000 exact
    const float* base = s + ((size_t)r * N_ENT << 6) + col;
    v2f a = {1.0f, 1.0f};
    v8f c = {};
    for (int n = n0; n < n0 + 400; n += 4) {
        v2f bm;
        bm.x = fabsf(base[(size_t)(n + kA)     << 6]);
        bm.y = fabsf(base[(size_t)(n + kA + 1) << 6]);
        c = __builtin_amdgcn_wmma_f32_16x16x4_f32(false, a, false, bm,
                                                  (short)0, c, false, false);
    }
    // c[0] = D[M=0][N=col%16] = column sum for this lane's column (lanes 0-15)
    if (lane < 16) atomAddF(&norm[(r << 6) + col], c[0]);
}

__global__ void k_scale(float* __restrict__ s, const float* __restrict__ norm) {
    int r = blockIdx.y;
    int i = blockIdx.x * blockDim.x + threadIdx.x;
    int b = i & 63;
    float nv = fmaxf(norm[(r << 6) + b], THRV);
    size_t idx = (size_t)r * PER_RANK + i;
    s[idx] = s[idx] / nv;
}

// out[b][n] = sum_r mem[r][n][b]
__global__ void k_finalize(float* __restrict__ out, const float* __restrict__ mem) {
    int b = blockIdx.y;
    int n = blockIdx.x * blockDim.x + threadIdx.x;
    if (n < N_ENT) {
        float acc = 0.0f;
#pragma unroll
        for (int r = 0; r < RANKS; ++r)
            acc += mem[((size_t)(r * N_ENT + n) << 6) + b];
        out[(size_t)b * N_ENT + n] = acc;
    }
}

extern "C" void kernel_launch(void* const* d_in, const int* in_sizes, int n_in,
                              void* d_out, int out_size, void* d_ws, size_t ws_size,
                              hipStream_t stream) {
    const float* att  = (const float*)d_in[0];   // [3,3,64,13]
    const int*   tt   = (const int*)  d_in[1];   // [64]
    const int*   rows = (const int*)  d_in[2];   // [12,100000]
    const int*   cols = (const int*)  d_in[3];   // [12,100000]
    const float* vals = (const float*)d_in[4];   // [12,100000]
    float* out = (float*)d_out;                  // [64,50000]

    float* bufA = (float*)d_ws;                  // [3][50000][64]
    float* bufB = bufA + TOTAL;                  // [3][50000][64]
    float* norm = bufB + TOTAL;                  // [3][64]

    k_zero<<<(TOTAL + 255) / 256, 256, 0, stream>>>(bufA, TOTAL);
    k_onehot<<<1, RANKS * BATCH, 0, stream>>>(bufA, tt);

    float* mem = bufA;
    float* s   = bufB;
    for (int step = 0; step < NSTEP - 1; ++step) {
        k_selfloop<<<dim3(PER_RANK / 256, RANKS), 256, 0, stream>>>(s, mem, att, step);
        k_edges<<<dim3(NNZE * BATCH / 256, NOP), 256, 0, stream>>>(s, mem, att,
                                                                   rows, cols, vals, step);
        k_zero<<<1, 256, 0, stream>>>(norm, RANKS * BATCH);
        k_norm_wmma<<<dim3(N_ENT / 400, RANKS), 128, 0, stream>>>(norm, s);
        k_scale<<<dim3(PER_RANK / 256, RANKS), 256, 0, stream>>>(s, norm);
        float* tmp = mem; mem = s; s = tmp;
    }
    k_finalize<<<dim3((N_ENT + 255) / 256, BATCH), 256, 0, stream>>>(out, mem);
}